// RotationPerturbationLayer_14448269984313
// MI455X (gfx1250) — compile-verified
//
#include <hip/hip_runtime.h>

typedef __attribute__((ext_vector_type(2))) float v2f;
typedef __attribute__((ext_vector_type(8))) float v8f;

static constexpr int Himg = 64;
static constexpr int Wimg = 64;
static constexpr int Npix = Himg * Wimg;   // 4096
static constexpr int Cch  = 3;
static constexpr int Bsz  = 8;

__device__ __forceinline__ float tent(float z) {
    // relu(1 - |z|)
    return fmaxf(0.0f, 1.0f - fabsf(z));
}

// One workgroup = (batch b, tile of 16 output pixels). 4 waves; each wave owns
// 3 j-tiles of the 192-wide (c,h) axis. Stage 1 via V_WMMA_F32_16X16X4_F32,
// stage 2 via shfl reduction + LDS float atomics.
__launch_bounds__(128)
__global__ void rot_perturb_wmma(const float* __restrict__ theta,
                                 const float* __restrict__ image,
                                 float* __restrict__ out) {
    __shared__ float acc_lds[Cch * 16];

    const int b      = blockIdx.x >> 8;    // 0..7
    const int ntile  = blockIdx.x & 255;   // 0..255
    const int n_base = ntile << 4;

    const int tid  = threadIdx.x;
    const int lane = tid & 31;
    const int wave = tid >> 5;
    const int m    = lane & 15;            // row (M) / column (N) index
    const int hv   = lane >> 4;            // half: K-group selector

    if (tid < Cch * 16) acc_lds[tid] = 0.0f;
    __syncthreads();

    const float th = theta[b];
    const float ct = cosf(th);
    const float st = sinf(th);

    // src_x for the A-matrix row this lane supplies (row m in both halves)
    float srcx;
    {
        const int n  = n_base + m;
        const float xr = (float)(n & 63) - 31.5f;
        const float yr = (float)(n >> 6) - 31.5f;
        srcx = ct * xr + st * yr + 31.5f;
    }

    // Precompute all 16 A fragments (K = 4*kk + 2*half + {0,1})
    v2f a_frag[16];
#pragma unroll
    for (int kk = 0; kk < 16; ++kk) {
        const float w0 = (float)(kk * 4 + hv * 2);
        a_frag[kk].x = tent(srcx - w0);
        a_frag[kk].y = tent(srcx - (w0 + 1.0f));
    }

    // src_y for the 8 D-fragment rows this lane's half holds (M = 8*half + r)
    float srcy_r[8];
#pragma unroll
    for (int r = 0; r < 8; ++r) {
        const int n  = n_base + hv * 8 + r;
        const float xr = (float)(n & 63) - 31.5f;
        const float yr = (float)(n >> 6) - 31.5f;
        srcy_r[r] = -st * xr + ct * yr + 31.5f;
    }

#pragma unroll
    for (int t = 0; t < 3; ++t) {
        const int jt = wave * 3 + t;       // 0..11, uniform per wave
        const int j  = jt * 16 + m;        // (c,h) column index, 0..191
        const int c  = j >> 6;             // uniform within the j-tile
        const int h  = j & 63;
        const float* brow = image + (c * Himg + h) * Wimg;  // image[c,h,:]

        v8f acc = {};
#pragma unroll
        for (int kk = 0; kk < 16; ++kk) {
            const int w = kk * 4 + hv * 2;             // even -> 8B aligned
            const v2f bf = *(const v2f*)(brow + w);    // B frag: K=2*half+{0,1}
            acc = __builtin_amdgcn_wmma_f32_16x16x4_f32(
                /*neg_a=*/false, a_frag[kk],
                /*neg_b=*/false, bf,
                /*c_mod=*/(short)0, acc,
                /*reuse_a=*/false, /*reuse_b=*/false);
        }

        // Stage 2: out[c, n_base+mm] += sum_j tent(src_y - h_j) * S1
#pragma unroll
        for (int r = 0; r < 8; ++r) {
            float contrib = acc[r] * tent(srcy_r[r] - (float)h);
            // reduce across the 16-lane group (same output element)
            contrib += __shfl_xor(contrib, 1, 32);
            contrib += __shfl_xor(contrib, 2, 32);
            contrib += __shfl_xor(contrib, 4, 32);
            contrib += __shfl_xor(contrib, 8, 32);
            if (m == 0) {
                const int mm = r + hv * 8;
                atomicAdd(&acc_lds[c * 16 + mm], contrib);  // ds_add_f32
            }
        }
    }

    __syncthreads();
    if (tid < Cch * 16) {
        const int c  = tid >> 4;
        const int mm = tid & 15;
        out[(b * Cch + c) * Npix + n_base + mm] = acc_lds[tid];
    }
}

extern "C" void kernel_launch(void* const* d_in, const int* in_sizes, int n_in,
                              void* d_out, int out_size, void* d_ws, size_t ws_size,
                              hipStream_t stream) {
    const float* theta = (const float*)d_in[0];   // [8,1] fp32
    const float* image = (const float*)d_in[1];   // [3,64,64] fp32
    float* out = (float*)d_out;                   // [8,3,64,64] fp32

    dim3 grid(Bsz * (Npix / 16));  // 2048 workgroups
    dim3 block(128);               // 4 waves
    hipLaunchKernelGGL(rot_perturb_wmma, grid, block, 0, stream,
                       theta, image, out);
}